// ShallowGCN_68358699483284
// MI455X (gfx1250) — compile-verified
//
#include <hip/hip_runtime.h>
#include <hip/hip_bf16.h>
#include <math.h>

#define N_NODES   100000
#define N_EDGES   1200000
#define N_FEAT    128
#define EMB       64
#define N_CLASSES 2
#define N_GRAPHS  256

typedef __attribute__((ext_vector_type(2))) float v2f;
typedef __attribute__((ext_vector_type(8))) float v8f;

// ---------------- degree / normalization ----------------

__global__ void k_init_deg(float* deg) {
    int i = blockIdx.x * blockDim.x + threadIdx.x;
    if (i < N_NODES) deg[i] = 1.0f;   // self-loop contributes 1 to every node
}

__global__ void k_accum_deg(const int* __restrict__ dst, float* deg) {
    int e = blockIdx.x * blockDim.x + threadIdx.x;
    if (e < N_EDGES) atomicAdd(&deg[dst[e]], 1.0f);
}

__global__ void k_deg_to_dinv(float* deg) {
    int i = blockIdx.x * blockDim.x + threadIdx.x;
    if (i < N_NODES) deg[i] = rsqrtf(deg[i]);   // deg >= 1 always (self-loops)
}

// ---------------- fp32 WMMA GEMM: H[N x EMB] = A[N x K] @ W[K x EMB] ----------------
// One wave computes a 16x16 output tile via V_WMMA_F32_16X16X4_F32.
// Block = 256 threads = 8 waves = 2 row-tiles x 4 col-tiles = 32 rows x 64 cols.

__global__ __launch_bounds__(256)
void k_gemm_wmma(const float* __restrict__ A, const float* __restrict__ W,
                 float* __restrict__ H, int K) {
    __shared__ float Wlds[N_FEAT * EMB];        // max 128*64 f32 = 32 KB
    for (int i = threadIdx.x; i < K * EMB; i += 256) Wlds[i] = W[i];
    __syncthreads();

    const int lane = threadIdx.x & 31;
    const int wave = threadIdx.x >> 5;
    const int col_base = (wave & 3) * 16;
    const int row_base = (blockIdx.x * 2 + (wave >> 2)) * 16;
    if (row_base >= N_NODES) return;

    const int m     = lane & 15;        // row (A,M) / col (B,N) within tile
    const int khalf = (lane >> 4) * 2;  // K sub-offset per half-wave

    v8f c = {};
    const float* arow = A + (size_t)(row_base + m) * K;
    for (int k0 = 0; k0 < K; k0 += 4) {
        v2f a = *(const v2f*)(arow + k0 + khalf);           // A[m][k0+khalf +0/+1]
        v2f b;
        b.x = Wlds[(k0 + khalf + 0) * EMB + col_base + m];  // B[k][n]
        b.y = Wlds[(k0 + khalf + 1) * EMB + col_base + m];
        c = __builtin_amdgcn_wmma_f32_16x16x4_f32(
                /*neg_a=*/false, a, /*neg_b=*/false, b,
                /*c_mod=*/(short)0, c, /*reuse_a=*/false, /*reuse_b=*/false);
    }

    float* out = H + (size_t)(row_base + (lane >> 4) * 8) * EMB + col_base + m;
#pragma unroll
    for (int r = 0; r < 8; ++r) out[(size_t)r * EMB] = c[r];
}

// ---------------- aggregation ----------------

// agg = bias + self-loop term (norm_ii = dinv[i]^2)
__global__ void k_init_agg(const float* __restrict__ h, const float* __restrict__ dinv,
                           const float* __restrict__ bias, float* __restrict__ agg) {
    int i = blockIdx.x * blockDim.x + threadIdx.x;
    if (i >= N_NODES * EMB) return;
    int n = i >> 6, f = i & 63;
    float d = dinv[n];
    agg[i] = bias[f] + d * d * h[i];
}

// 64 consecutive threads handle one edge: coalesced 256B gather + coalesced atomics (L2-resident)
__global__ void k_scatter(const int* __restrict__ src, const int* __restrict__ dst,
                          const float* __restrict__ dinv, const float* __restrict__ h,
                          float* __restrict__ agg) {
    unsigned int t = blockIdx.x * blockDim.x + threadIdx.x;
    if (t >= (unsigned int)N_EDGES * EMB) return;
    int e = (int)(t >> 6), f = (int)(t & 63);
    int s = src[e], d = dst[e];
    float w = dinv[s] * dinv[d];
    atomicAdd(&agg[(size_t)d * EMB + f], w * h[(size_t)s * EMB + f]);
}

__global__ void k_tanh(const float* __restrict__ agg, float* __restrict__ act) {
    int i = blockIdx.x * blockDim.x + threadIdx.x;
    if (i < N_NODES * EMB) act[i] = tanhf(agg[i]);
}

// ---------------- pooling + head ----------------

__device__ __forceinline__ void atomicMaxF(float* addr, float val) {
    if (val >= 0.0f) atomicMax((int*)addr, __float_as_int(val));
    else             atomicMin((unsigned int*)addr, __float_as_uint(val));
}

__global__ void k_pool_init(float* gsum, float* gmax, float* counts) {
    int i = blockIdx.x * blockDim.x + threadIdx.x;
    if (i < N_GRAPHS * EMB) {
        gsum[i] = 0.0f;
        ((unsigned int*)gmax)[i] = 0xFF800000u;   // -inf
    }
    if (i < N_GRAPHS) counts[i] = 0.0f;
}

__global__ void k_pool(const int* __restrict__ batch, const float* __restrict__ act,
                       float* gsum, float* gmax, float* counts) {
    int i = blockIdx.x * blockDim.x + threadIdx.x;
    if (i >= N_NODES * EMB) return;
    int n = i >> 6, f = i & 63;
    int g = batch[n];
    float v = act[i];
    atomicAdd(&gsum[g * EMB + f], v);
    atomicMaxF(&gmax[g * EMB + f], v);
    if (f == 0) atomicAdd(&counts[g], 1.0f);
}

__global__ void k_head(const float* __restrict__ gsum, const float* __restrict__ gmax,
                       const float* __restrict__ counts, const float* __restrict__ Wout,
                       const float* __restrict__ bout, float* __restrict__ out) {
    int g = blockIdx.x * blockDim.x + threadIdx.x;
    if (g >= N_GRAPHS) return;
    float inv_cnt = 1.0f / fmaxf(counts[g], 1.0f);
    float acc0 = bout[0], acc1 = bout[1];
    for (int f = 0; f < EMB; ++f) {
        float mx = gmax[g * EMB + f];
        float mn = gsum[g * EMB + f] * inv_cnt;
        acc0 += mx * Wout[f * 2 + 0] + mn * Wout[(EMB + f) * 2 + 0];
        acc1 += mx * Wout[f * 2 + 1] + mn * Wout[(EMB + f) * 2 + 1];
    }
    out[g * 2 + 0] = acc0;
    out[g * 2 + 1] = acc1;
}

// ---------------- launch ----------------

extern "C" void kernel_launch(void* const* d_in, const int* in_sizes, int n_in,
                              void* d_out, int out_size, void* d_ws, size_t ws_size,
                              hipStream_t stream) {
    (void)in_sizes; (void)n_in; (void)out_size; (void)ws_size;

    const float* x     = (const float*)d_in[0];
    const int*   eidx  = (const int*)d_in[1];
    const int*   src   = eidx;            // edge_index[0]
    const int*   dst   = eidx + N_EDGES;  // edge_index[1]
    const int*   batch = (const int*)d_in[2];
    const float* W[4]  = { (const float*)d_in[3], (const float*)d_in[5],
                           (const float*)d_in[7], (const float*)d_in[9] };
    const float* B[4]  = { (const float*)d_in[4], (const float*)d_in[6],
                           (const float*)d_in[8], (const float*)d_in[10] };
    const float* Wout  = (const float*)d_in[11];
    const float* bout  = (const float*)d_in[12];
    float* out = (float*)d_out;

    // workspace carve (floats): ~77.3 MB total
    float* ws     = (float*)d_ws;
    float* dinv   = ws;                                 // N
    float* h      = dinv + N_NODES;                     // N*EMB
    float* agg    = h + (size_t)N_NODES * EMB;          // N*EMB
    float* act    = agg + (size_t)N_NODES * EMB;        // N*EMB
    float* gsum   = act + (size_t)N_NODES * EMB;        // G*EMB
    float* gmax   = gsum + N_GRAPHS * EMB;              // G*EMB
    float* counts = gmax + N_GRAPHS * EMB;              // G

    const int T = 256;
    const int gN   = (N_NODES + T - 1) / T;             // 391
    const int gE   = (N_EDGES + T - 1) / T;             // 4688
    const int gNF  = (N_NODES * EMB) / T;               // 25000
    const int gEF  = (N_EDGES * EMB) / T;               // 300000
    const int gMM  = N_NODES / 32;                      // 3125 (100000 % 32 == 0)

    // symmetric normalization: dinv = rsqrt(deg), deg includes self-loops
    k_init_deg   <<<gN, T, 0, stream>>>(dinv);
    k_accum_deg  <<<gE, T, 0, stream>>>(dst, dinv);
    k_deg_to_dinv<<<gN, T, 0, stream>>>(dinv);

    // 4 GCN layers
    for (int l = 0; l < 4; ++l) {
        const float* in = (l == 0) ? x : act;
        const int    K  = (l == 0) ? N_FEAT : EMB;
        k_gemm_wmma<<<gMM, T, 0, stream>>>(in, W[l], h, K);
        k_init_agg <<<gNF, T, 0, stream>>>(h, dinv, B[l], agg);
        k_scatter  <<<gEF, T, 0, stream>>>(src, dst, dinv, h, agg);
        k_tanh     <<<gNF, T, 0, stream>>>(agg, act);
    }

    // pooling + head
    k_pool_init<<<(N_GRAPHS * EMB + T - 1) / T, T, 0, stream>>>(gsum, gmax, counts);
    k_pool     <<<gNF, T, 0, stream>>>(batch, act, gsum, gmax, counts);
    k_head     <<<(N_GRAPHS + T - 1) / T, T, 0, stream>>>(gsum, gmax, counts, Wout, bout, out);
}